// ConstraintsModule_1451698946374
// MI455X (gfx1250) — compile-verified
//
#include <hip/hip_runtime.h>
#include <stdint.h>

// Problem dims (fixed by the reference).
#define BB   128   // batch
#define CC   1024  // clauses
#define NN   512   // atoms
#define NCLS 1024  // num classes
#define CT   4     // clauses per block in phase 1
#define NCHUNK 128 // n-tile per LDS buffer
#define NK   (NN / NCHUNK) // 4 chunks

#define AS1 __attribute__((address_space(1)))
#define AS3 __attribute__((address_space(3)))
typedef __attribute__((__vector_size__(4 * sizeof(int)))) int v4i_;

// ---- gfx1250 async-copy-to-LDS path (guarded; falls back to plain copy) ----
#if defined(__HIP_DEVICE_COMPILE__)
#if defined(__has_builtin)
#if __has_builtin(__builtin_amdgcn_global_load_async_to_lds_b128)
#define USE_ASYNC_LDS 1
#endif
#endif
#endif

__device__ __forceinline__ void async_cp_b128(const float* g, float* lds) {
#ifdef USE_ASYNC_LDS
  __builtin_amdgcn_global_load_async_to_lds_b128(
      (AS1 v4i_*)g,   // global source (AS1 int4*, per clang diagnostic)
      (AS3 v4i_*)lds, // LDS destination
      /*offset=*/0, /*cpol=*/0);
#else
  float4 v = *(const float4*)g;
  *(float4*)lds = v;
#endif
}

__device__ __forceinline__ void wait_async_le2() {
#ifdef USE_ASYNC_LDS
#if __has_builtin(__builtin_amdgcn_s_wait_asynccnt)
  __builtin_amdgcn_s_wait_asynccnt(2);
#else
  asm volatile("s_wait_asynccnt 2" ::: "memory");
#endif
#endif
}

__device__ __forceinline__ void wait_async_le0() {
#ifdef USE_ASYNC_LDS
#if __has_builtin(__builtin_amdgcn_s_wait_asynccnt)
  __builtin_amdgcn_s_wait_asynccnt(0);
#else
  asm volatile("s_wait_asynccnt 0" ::: "memory");
#endif
#endif
}

// ---- K0a: zero lb/ub accumulators, gather m transposed: m_t[n][b] ----------
__global__ __launch_bounds__(256) void k_init(const float* __restrict__ preds,
                                              const long long* __restrict__ atoms,
                                              float* __restrict__ mt,
                                              unsigned* __restrict__ lbbits,
                                              unsigned* __restrict__ ubbits) {
  int i = blockIdx.x * blockDim.x + threadIdx.x; // i = n*BB + b, i < BB*NN
  if (i >= BB * NN) return;
  lbbits[i] = 0u;
  ubbits[i] = 0u;
  int n = i >> 7;        // / BB
  int b = i & (BB - 1);  // % BB
  int a = (int)atoms[n];
  mt[i] = preds[b * NCLS + a];
}

// ---- K0b: per-clause head extraction (one-hot rows) ------------------------
__global__ __launch_bounds__(NN) void k_heads(const float* __restrict__ pos_head,
                                              const float* __restrict__ neg_head,
                                              int* __restrict__ head_idx,
                                              int* __restrict__ head_sign) {
  int c = blockIdx.x;
  int t = threadIdx.x; // 0..NN-1, coalesced row scan
  float ph = pos_head[(size_t)c * NN + t];
  float nh = neg_head[(size_t)c * NN + t];
  if (ph != 0.0f) { head_idx[c] = t; head_sign[c] = 1; }
  if (nh != 0.0f) { head_idx[c] = t; head_sign[c] = 0; }
}

// ---- K1: body_min[c][b] via async double-buffered LDS tiles ----------------
// Block = 128 threads (one per batch b), handles CT=4 clauses, loops NK chunks
// of 128 atoms. Clause rows stream into LDS with ASYNCcnt-tracked copies while
// the previous chunk is max-fma reduced.
__global__ __launch_bounds__(BB) void k_bodymin(const float* __restrict__ pos_body,
                                                const float* __restrict__ neg_body,
                                                const float* __restrict__ mt,
                                                float* __restrict__ bodymin) {
  __shared__ __align__(16) float sp[2][CT][NCHUNK];
  __shared__ __align__(16) float sn[2][CT][NCHUNK];

  const int c0 = blockIdx.x * CT;
  const int t  = threadIdx.x;      // batch index b
  const int jr = t >> 5;           // which clause row this thread helps fill
  const int lr = t & 31;           // 16B segment within the 512B row chunk

  const float* gp = pos_body + (size_t)(c0 + jr) * NN + lr * 4;
  const float* gn = neg_body + (size_t)(c0 + jr) * NN + lr * 4;

  // Prologue: chunk 0 -> buffer 0 (2 async ops outstanding per wave).
  async_cp_b128(gp, &sp[0][jr][lr * 4]);
  async_cp_b128(gn, &sn[0][jr][lr * 4]);

  float acc[CT];
#pragma unroll
  for (int j = 0; j < CT; ++j) acc[j] = 0.0f;

  for (int k = 0; k < NK; ++k) {
    const int buf = k & 1;
    if (k + 1 < NK) {
      const int nb = (k + 1) & 1; // buffer last touched at chunk k-1; barrier-protected
      async_cp_b128(gp + (k + 1) * NCHUNK, &sp[nb][jr][lr * 4]);
      async_cp_b128(gn + (k + 1) * NCHUNK, &sn[nb][jr][lr * 4]);
      wait_async_le2(); // 4 outstanding, in-order completion => chunk k landed
    } else {
      wait_async_le0();
    }
    __syncthreads(); // all waves' tile data visible

    const float* mrow = mt + (size_t)(k * NCHUNK) * BB + t; // coalesced, L2-resident
    for (int n = 0; n < NCHUNK; ++n) {
      float v = mrow[(size_t)n * BB];
#pragma unroll
      for (int j = 0; j < CT; ++j) {
        float p = sp[buf][j][n];             // LDS broadcast (uniform addr)
        float q = sn[buf][j][n];
        acc[j] = fmaxf(acc[j], fmaf(v, q - p, p)); // body_rev, exact for p,q in {0,1}
      }
    }
    __syncthreads(); // protect buffer reuse
  }

#pragma unroll
  for (int j = 0; j < CT; ++j)
    bodymin[(size_t)(c0 + j) * BB + t] = 1.0f - acc[j]; // coalesced plain store
}

// ---- K2: scatter-max body_min into lb / ub via bitwise atomic max ----------
// All values are in [0,1] (>= 0), so uint bit-pattern max == float max.
__global__ __launch_bounds__(BB) void k_scatter(const float* __restrict__ bodymin,
                                                const int* __restrict__ head_idx,
                                                const int* __restrict__ head_sign,
                                                unsigned* __restrict__ lbbits,
                                                unsigned* __restrict__ ubbits) {
  int c = blockIdx.x;
  int b = threadIdx.x;
  float bm = bodymin[(size_t)c * BB + b];
  int j = head_idx[c];
  unsigned bits = __float_as_uint(bm);
  unsigned* dst = head_sign[c] ? lbbits : ubbits;
  atomicMax(&dst[(size_t)b * NN + j], bits);
}

// ---- K3: out = preds (columns not in atoms pass through) -------------------
__global__ __launch_bounds__(256) void k_copy(const float* __restrict__ preds,
                                              float* __restrict__ out) {
  int i = blockIdx.x * blockDim.x + threadIdx.x;
  if (i < BB * NCLS) out[i] = preds[i];
}

// ---- K4: clamp update and scatter into the atom columns --------------------
__global__ __launch_bounds__(256) void k_final(const float* __restrict__ preds,
                                               const long long* __restrict__ atoms,
                                               const unsigned* __restrict__ lbbits,
                                               const unsigned* __restrict__ ubbits,
                                               float* __restrict__ out) {
  int i = blockIdx.x * blockDim.x + threadIdx.x; // i = b*NN + n
  if (i >= BB * NN) return;
  int b = i / NN;
  int n = i - b * NN;
  int a = (int)atoms[n];
  float m  = preds[b * NCLS + a];
  float lb = __uint_as_float(lbbits[i]);
  float ub = 1.0f - __uint_as_float(ubbits[i]);
  float lo = fminf(lb, ub);
  float hi = fmaxf(lb, ub);
  out[b * NCLS + a] = fmaxf(lo, fminf(hi, m));
}

extern "C" void kernel_launch(void* const* d_in, const int* in_sizes, int n_in,
                              void* d_out, int out_size, void* d_ws, size_t ws_size,
                              hipStream_t stream) {
  (void)in_sizes; (void)n_in; (void)out_size; (void)ws_size;
  const float*     preds    = (const float*)d_in[0];
  const float*     pos_head = (const float*)d_in[1];
  const float*     neg_head = (const float*)d_in[2];
  const float*     pos_body = (const float*)d_in[3];
  const float*     neg_body = (const float*)d_in[4];
  const long long* atoms    = (const long long*)d_in[5]; // int64
  float*           out      = (float*)d_out;

  // Workspace layout (~1.26 MB total)
  float*    mt       = (float*)d_ws;                       // NN*BB
  float*    bodymin  = mt + (size_t)NN * BB;               // CC*BB
  unsigned* lbbits   = (unsigned*)(bodymin + (size_t)CC * BB); // BB*NN
  unsigned* ubbits   = lbbits + (size_t)BB * NN;           // BB*NN
  int*      head_idx = (int*)(ubbits + (size_t)BB * NN);   // CC
  int*      head_sig = head_idx + CC;                      // CC

  k_init   <<<(BB * NN + 255) / 256, 256, 0, stream>>>(preds, atoms, mt, lbbits, ubbits);
  k_heads  <<<CC, NN, 0, stream>>>(pos_head, neg_head, head_idx, head_sig);
  k_bodymin<<<CC / CT, BB, 0, stream>>>(pos_body, neg_body, mt, bodymin);
  k_scatter<<<CC, BB, 0, stream>>>(bodymin, head_idx, head_sig, lbbits, ubbits);
  k_copy   <<<(BB * NCLS + 255) / 256, 256, 0, stream>>>(preds, out);
  k_final  <<<(BB * NN + 255) / 256, 256, 0, stream>>>(preds, atoms, lbbits, ubbits, out);
}